// Graph_Survival_Analysis_36782099923560
// MI455X (gfx1250) — compile-verified
//
#include <hip/hip_runtime.h>
#include <hip/hip_bf16.h>
#include <math.h>

typedef __attribute__((ext_vector_type(2))) float v2f;
typedef __attribute__((ext_vector_type(8))) float v8f;

#define NROWS 6144
#define EPS_DBL 2.220446049250313e-16f   // float(np.finfo(float).eps)
#define GNW 8                            // waves per block for split kernels

// ---------------------------------------------------------------------------
// CDNA5 fp32 WMMA: D(16x16 f32) = A(16x4 f32) * B(4x16 f32) + C
// A/B per-lane: lane<16 -> K = k0+{0,1}, lane>=16 -> K = k0+{2,3}
// C/D per-lane: vgpr r -> row (r + 8*(lane>=16)), col (lane&15)
// ---------------------------------------------------------------------------
__device__ __forceinline__ v8f wmma4(v2f a, v2f b, v8f c) {
  return __builtin_amdgcn_wmma_f32_16x16x4_f32(
      /*neg_a=*/false, a, /*neg_b=*/false, b,
      /*c_mod=*/(short)0, c, /*reuse_a=*/false, /*reuse_b=*/false);
}

// ===========================================================================
// Kernel 1: out[N x Mout] = act(A[N x K] @ W[Mout x K]^T + bias), Mout <= 48,
// K % 4 == 0. 8 waves per 16-row tile: each wave accumulates a disjoint
// K-slice with fp32 WMMA, partials reduced through LDS. ACT: 0=id, 1=tanh.
// ===========================================================================
template <int ACT>
__global__ __launch_bounds__(32 * GNW) void gemm_bias_act(
    const float* __restrict__ A, const float* __restrict__ W,
    const float* __restrict__ bias, float* __restrict__ out, int K, int Mout) {
  __shared__ __align__(16) float sAcc[GNW][16][49];
  const int tid = threadIdx.x;
  const int w = tid >> 5;
  const int lane = tid & 31;
  const int half = lane >> 4;
  const int lid = lane & 15;
  const int i0 = blockIdx.x * 16;

  const int steps = K >> 2;                 // K/4 WMMA steps total
  const int chunk = (steps + GNW - 1) / GNW;
  const int s0 = w * chunk;
  const int s1 = (s0 + chunk < steps) ? (s0 + chunk) : steps;

  const float* __restrict__ arow = A + (size_t)(i0 + lid) * K + 2 * half;
  const float* __restrict__ w0 = W + (size_t)(lid)*K + 2 * half;
  const float* __restrict__ w1 = W + (size_t)(16 + lid) * K + 2 * half;
  const bool c2ok = (32 + lid) < Mout;
  const float* __restrict__ w2 = W + (size_t)(c2ok ? (32 + lid) : 0) * K + 2 * half;

  v8f acc0 = {0.f, 0.f, 0.f, 0.f, 0.f, 0.f, 0.f, 0.f};
  v8f acc1 = acc0, acc2 = acc0;

  for (int s = s0; s < s1; ++s) {
    const int k0 = 4 * s;
    if (((s - s0) & 63) == 0 && (k0 + 1024) < K)
      __builtin_prefetch(arow + k0 + 1024, 0, 1);  // global_prefetch_b8
    v2f a = *(const v2f*)(arow + k0);
    v2f b0 = *(const v2f*)(w0 + k0);
    v2f b1 = *(const v2f*)(w1 + k0);
    v2f b2 = *(const v2f*)(w2 + k0);
    if (!c2ok) { b2.x = 0.f; b2.y = 0.f; }
    acc0 = wmma4(a, b0, acc0);
    acc1 = wmma4(a, b1, acc1);
    acc2 = wmma4(a, b2, acc2);
  }

  // dump per-wave partial tiles to LDS
#pragma unroll
  for (int r = 0; r < 8; ++r) {
    sAcc[w][r + 8 * half][lid] = acc0[r];
    sAcc[w][r + 8 * half][16 + lid] = acc1[r];
    sAcc[w][r + 8 * half][32 + lid] = acc2[r];
  }
  __syncthreads();

  // reduce 8 partials, add bias, activate, store
  for (int t = tid; t < 16 * 48; t += 32 * GNW) {
    int r = t / 48, c = t % 48;
    if (c < Mout) {
      float s = 0.f;
#pragma unroll
      for (int ww = 0; ww < GNW; ++ww) s += sAcc[ww][r][c];
      s += bias[c];
      if (ACT == 1) s = tanhf(s);
      out[(size_t)(i0 + r) * Mout + c] = s;
    }
  }
}

// ===========================================================================
// Kernel 2: 5-layer highway net, in-place on x[N x 40]. One wave / 16 rows.
// Per layer: 3 GEMMs [16x40]x[40x40] via WMMA (gate/nonlinear/linear) + gate.
// (~0.3 GFLOP total -> latency-trivial; occupancy not critical here.)
// ===========================================================================
__global__ __launch_bounds__(32) void highway_kernel(
    float* __restrict__ x, const float* __restrict__ hwW,
    const float* __restrict__ hwb) {
  __shared__ __align__(16) float sx[16][48];
  const int lane = threadIdx.x & 31;
  const int half = lane >> 4;
  const int lid = lane & 15;
  const int i0 = blockIdx.x * 16;

  for (int t = lane; t < 16 * 40; t += 32) {
    int r = t / 40, c = t % 40;
    sx[r][c] = x[(size_t)(i0 + r) * 40 + c];
  }
  __syncthreads();

  for (int layer = 0; layer < 5; ++layer) {
    const float* Wnl = hwW + (size_t)(0 * 5 + layer) * 1600;
    const float* Wln = hwW + (size_t)(1 * 5 + layer) * 1600;
    const float* Wg  = hwW + (size_t)(2 * 5 + layer) * 1600;
    const float* bnl = hwb + (0 * 5 + layer) * 40;
    const float* bln = hwb + (1 * 5 + layer) * 40;
    const float* bg  = hwb + (2 * 5 + layer) * 40;

    v8f z = {0.f, 0.f, 0.f, 0.f, 0.f, 0.f, 0.f, 0.f};
    v8f aN[3] = {z, z, z}, aL[3] = {z, z, z}, aG[3] = {z, z, z};

#pragma unroll
    for (int s = 0; s < 10; ++s) {
      const int k0 = 4 * s;
      v2f a = *(const v2f*)&sx[lid][k0 + 2 * half];
#pragma unroll
      for (int t = 0; t < 3; ++t) {
        int col = t * 16 + lid;
        v2f bN, bL, bG2;
        if (col < 40) {
          bN  = *(const v2f*)(Wnl + (size_t)col * 40 + k0 + 2 * half);
          bL  = *(const v2f*)(Wln + (size_t)col * 40 + k0 + 2 * half);
          bG2 = *(const v2f*)(Wg  + (size_t)col * 40 + k0 + 2 * half);
        } else {
          bN.x = bN.y = bL.x = bL.y = bG2.x = bG2.y = 0.f;
        }
        aN[t] = wmma4(a, bN, aN[t]);
        aL[t] = wmma4(a, bL, aL[t]);
        aG[t] = wmma4(a, bG2, aG[t]);
      }
    }
    __syncthreads();
#pragma unroll
    for (int t = 0; t < 3; ++t) {
      int col = t * 16 + lid;
      if (col < 40) {
        float vbn = bnl[col], vbl = bln[col], vbg = bg[col];
#pragma unroll
        for (int r = 0; r < 8; ++r) {
          float g  = 1.f / (1.f + __expf(-(aG[t][r] + vbg)));
          float nl = fmaxf(aN[t][r] + vbn, 0.f);
          float ln = fmaxf(aL[t][r] + vbl, 0.f);
          sx[r + 8 * half][col] = g * nl + (1.f - g) * ln;
        }
      }
    }
    __syncthreads();
  }

  for (int t = lane; t < 16 * 40; t += 32) {
    int r = t / 40, c = t % 40;
    x[(size_t)(i0 + r) * 40 + c] = sx[r][c];
  }
}

// ===========================================================================
// Kernel 3: fused features + L2-normalized (zero-padded to 64) feature sets.
// ===========================================================================
template <int D>
__device__ __forceinline__ void norm_emit(const float* v, float* __restrict__ feat,
                                          float* __restrict__ sqv, int i) {
  float s = 0.f;
#pragma unroll
  for (int c = 0; c < D; ++c) s += v[c] * v[c];
  float inv = 1.f / fmaxf(sqrtf(s), 1e-12f);
  float s2 = 0.f;
#pragma unroll
  for (int c = 0; c < 64; ++c) {
    float xv = (c < D) ? v[c] * inv : 0.f;
    feat[(size_t)i * 64 + c] = xv;
    s2 += xv * xv;
  }
  sqv[i] = s2;
}

__global__ void prep_kernel(const float* __restrict__ rna, const float* __restrict__ mi,
                            const float* __restrict__ rlin, const float* __restrict__ mlin,
                            float* __restrict__ fused, float* __restrict__ featG,
                            float* __restrict__ featM, float* __restrict__ featF,
                            float* __restrict__ sqG, float* __restrict__ sqM,
                            float* __restrict__ sqF, int N) {
  int i = blockIdx.x * blockDim.x + threadIdx.x;
  if (i >= N) return;
  float r[40], m[40], f[60];
#pragma unroll
  for (int c = 0; c < 40; ++c) {
    r[c] = rna[(size_t)i * 40 + c];
    m[c] = mi[(size_t)i * 40 + c];
    f[c] = r[c] + m[c];
  }
#pragma unroll
  for (int o = 0; o < 20; ++o) {
    float a = 0.f, b = 0.f;
#pragma unroll
    for (int c = 0; c < 40; ++c) {
      a += rlin[o * 40 + c] * r[c];
      b += mlin[o * 40 + c] * m[c];
    }
    f[40 + o] = a * b;
  }
#pragma unroll
  for (int c = 0; c < 60; ++c) fused[(size_t)i * 60 + c] = f[c];
  norm_emit<40>(r, featG, sqG, i);
  norm_emit<40>(m, featM, sqM, i);
  norm_emit<60>(f, featF, sqF, i);
}

// ===========================================================================
// Kernel 4: gram-matrix top-k (heavy WMMA stage). 8 waves per 16-row tile;
// each wave scans an interleaved 1/8 of the column tiles with a private
// per-row top-k (registers), partial lists merged per row through LDS with
// (value desc, index asc) comparator == lax.top_k tie behavior.
// ===========================================================================
template <int KSEL>
__global__ __launch_bounds__(32 * GNW) void gram_topk(
    const float* __restrict__ feat, const float* __restrict__ sq,
    int* __restrict__ idx_out, int N) {
  __shared__ __align__(16) float sTile[GNW][16][17];
  __shared__ float sSq[GNW][16];
  __shared__ float sVals[GNW][16][KSEL];
  __shared__ int sIds[GNW][16][KSEL];

  const int tid = threadIdx.x;
  const int w = tid >> 5;
  const int lane = tid & 31;
  const int half = lane >> 4;
  const int lid = lane & 15;
  const int i0 = blockIdx.x * 16;

  // cache this row-tile's A fragments for all 16 K-steps
  v2f aF[16];
  const float* myrow = feat + (size_t)(i0 + lid) * 64 + 2 * half;
#pragma unroll
  for (int s = 0; s < 16; ++s) aF[s] = *(const v2f*)(myrow + 4 * s);
  const float sqi = sq[i0 + lid];

  float vals[KSEL];
  int ids[KSEL];
#pragma unroll
  for (int t = 0; t < KSEL; ++t) { vals[t] = -1e30f; ids[t] = 0x7fffffff; }

  const int iters = N / (16 * GNW);  // uniform trip count for all waves
  for (int it = 0; it < iters; ++it) {
    const int j0 = (it * GNW + w) * 16;
    const float* jrow = feat + (size_t)(j0 + lid) * 64 + 2 * half;
    v8f acc = {0.f, 0.f, 0.f, 0.f, 0.f, 0.f, 0.f, 0.f};
#pragma unroll
    for (int s = 0; s < 16; ++s) {
      v2f bF = *(const v2f*)(jrow + 4 * s);
      acc = wmma4(aF[s], bF, acc);
    }
    __syncthreads();  // prior iteration's sTile reads complete
#pragma unroll
    for (int r = 0; r < 8; ++r) sTile[w][r + 8 * half][lid] = acc[r];
    if (half == 0) sSq[w][lid] = sq[j0 + lid];
    __syncthreads();
    if (half == 0) {
#pragma unroll
      for (int c = 0; c < 16; ++c) {
        float dotv = sTile[w][lid][c];
        float d2 = fmaxf(sqi + sSq[w][c] - 2.f * dotv, 0.f);
        float v = -d2;  // monotone with W = exp(-d2/20)
        int id = j0 + c;
#pragma unroll
        for (int p = 0; p < KSEL; ++p) {  // stable descending bubble-insert
          if (v > vals[p]) {
            float tv = vals[p]; int ti = ids[p];
            vals[p] = v; ids[p] = id;
            v = tv; id = ti;
          }
        }
      }
    }
  }

  // publish per-wave partial lists and merge
  if (half == 0) {
#pragma unroll
    for (int t = 0; t < KSEL; ++t) {
      sVals[w][lid][t] = vals[t];
      sIds[w][lid][t] = ids[t];
    }
  }
  __syncthreads();
  if (tid < 16) {
    float mv[KSEL];
    int mi_[KSEL];
#pragma unroll
    for (int t = 0; t < KSEL; ++t) { mv[t] = -1e30f; mi_[t] = 0x7fffffff; }
#pragma unroll
    for (int ww = 0; ww < GNW; ++ww) {
#pragma unroll
      for (int t = 0; t < KSEL; ++t) {
        float v = sVals[ww][tid][t];
        int id = sIds[ww][tid][t];
#pragma unroll
        for (int p = 0; p < KSEL; ++p) {
          bool better = (v > mv[p]) || (v == mv[p] && id < mi_[p]);
          if (better) {
            float tv = mv[p]; int ti = mi_[p];
            mv[p] = v; mi_[p] = id;
            v = tv; id = ti;
          }
        }
      }
    }
#pragma unroll
    for (int t = 0; t < KSEL; ++t)
      idx_out[(size_t)(i0 + tid) * KSEL + t] = mi_[t];
  }
}

// ===========================================================================
// Kernel 5: mutual-kNN sparse weights. Masked W is symmetric -> colsum==rowsum.
// ===========================================================================
template <int KSEL>
__global__ void sparse_w(const float* __restrict__ feat, const float* __restrict__ sq,
                         const int* __restrict__ idx, float* __restrict__ wval,
                         float* __restrict__ rowsum, int N) {
  int i = blockIdx.x * blockDim.x + threadIdx.x;
  if (i >= N) return;
  float sqi = sq[i];
  float rs = 0.f;
#pragma unroll
  for (int t = 0; t < KSEL; ++t) {
    int j = idx[(size_t)i * KSEL + t];
    bool mutual = false;
#pragma unroll
    for (int u = 0; u < KSEL; ++u) mutual |= (idx[(size_t)j * KSEL + u] == i);
    float w = 0.f;
    if (mutual) {
      float dot = 0.f;
#pragma unroll
      for (int c = 0; c < 64; ++c)
        dot += feat[(size_t)i * 64 + c] * feat[(size_t)j * 64 + c];
      float d2 = fmaxf(sqi + sq[j] - 2.f * dot, 0.f);
      w = __expf(-d2 * 0.05f);  // exp(-d2/20)
    }
    wval[(size_t)i * KSEL + t] = w;
    rs += w;
  }
  rowsum[i] = rs;
}

// ===========================================================================
// Kernel 6: row sums of Wt -> sinv[i] = 1/(rowsumWt[i]+EPS) = d[i]^2.
// ===========================================================================
__global__ void degree_kernel(const int* __restrict__ iG, const float* __restrict__ wG,
                              const float* __restrict__ rG, const int* __restrict__ iF,
                              const float* __restrict__ wF, const float* __restrict__ rF,
                              const int* __restrict__ iM, const float* __restrict__ wM,
                              const float* __restrict__ rM, float* __restrict__ sinv,
                              int N) {
  int i = blockIdx.x * blockDim.x + threadIdx.x;
  if (i >= N) return;
  float s = 1.f;  // identity
#pragma unroll
  for (int t = 0; t < 5; ++t) { int j = iG[(size_t)i * 5 + t]; s += wG[(size_t)i * 5 + t] / rG[j]; }
#pragma unroll
  for (int t = 0; t < 3; ++t) { int j = iF[(size_t)i * 3 + t]; s += wF[(size_t)i * 3 + t] / rF[j]; }
#pragma unroll
  for (int t = 0; t < 3; ++t) { int j = iM[(size_t)i * 3 + t]; s += wM[(size_t)i * 3 + t] / rM[j]; }
  sinv[i] = 1.f / (s + EPS_DBL);
}

// ===========================================================================
// Kernel 7: batchnorm on fused -> xb [N x 60]
// ===========================================================================
__global__ void bn_kernel(const float* __restrict__ fused, const float* __restrict__ gamma,
                          const float* __restrict__ beta, const float* __restrict__ mean,
                          const float* __restrict__ var, float* __restrict__ xb, int total) {
  int t = blockIdx.x * blockDim.x + threadIdx.x;
  if (t >= total) return;
  int c = t % 60;
  xb[t] = (fused[t] - mean[c]) * rsqrtf(var[c] + 1e-5f) * gamma[c] + beta[c];
}

// ===========================================================================
// Kernel 8: sparse S@X + cox head. S[i,j] = sinv[j] * Wt[i,j] (faithful quirk:
// columns scaled twice). Wt row has <=11 nnz + identity.
// ===========================================================================
template <int KS>
__device__ __forceinline__ void accum_graph(int i, const int* __restrict__ idx,
                                            const float* __restrict__ wv,
                                            const float* __restrict__ rs,
                                            const float* __restrict__ sinv,
                                            const float* __restrict__ X, float* Y) {
#pragma unroll
  for (int t = 0; t < KS; ++t) {
    int j = idx[(size_t)i * KS + t];
    float w = wv[(size_t)i * KS + t];
    if (w > 0.f) {
      float sc = sinv[j] * w / rs[j];
#pragma unroll
      for (int c = 0; c < 40; ++c) Y[c] += sc * X[(size_t)j * 40 + c];
    }
  }
}

__global__ void final_kernel(const float* __restrict__ X, const float* __restrict__ sinv,
                             const int* __restrict__ iG, const float* __restrict__ wG,
                             const float* __restrict__ rG, const int* __restrict__ iF,
                             const float* __restrict__ wF, const float* __restrict__ rF,
                             const int* __restrict__ iM, const float* __restrict__ wM,
                             const float* __restrict__ rM, const float* __restrict__ c1W,
                             const float* __restrict__ c1b, const float* __restrict__ c2W,
                             const float* __restrict__ c2b, float* __restrict__ out, int N) {
  int i = blockIdx.x * blockDim.x + threadIdx.x;
  if (i >= N) return;
  float Y[40];
  float si = sinv[i];
#pragma unroll
  for (int c = 0; c < 40; ++c) Y[c] = si * X[(size_t)i * 40 + c];  // identity term
  accum_graph<5>(i, iG, wG, rG, sinv, X, Y);
  accum_graph<3>(i, iF, wF, rF, sinv, X, Y);
  accum_graph<3>(i, iM, wM, rM, sinv, X, Y);
  float o = c2b[0];
#pragma unroll
  for (int m = 0; m < 40; ++m) {
    float a = c1b[m];
#pragma unroll
    for (int c = 0; c < 40; ++c) a += c1W[m * 40 + c] * Y[c];
    o += fmaxf(a, 0.f) * c2W[m];
  }
  out[i] = o;
}

// ===========================================================================
extern "C" void kernel_launch(void* const* d_in, const int* in_sizes, int n_in,
                              void* d_out, int out_size, void* d_ws, size_t ws_size,
                              hipStream_t stream) {
  (void)in_sizes; (void)n_in; (void)out_size; (void)ws_size;
  const int N = NROWS;

  const float* gene   = (const float*)d_in[0];
  const float* miRNA  = (const float*)d_in[1];
  const float* rna_W  = (const float*)d_in[2];
  const float* rna_b  = (const float*)d_in[3];
  const float* rna_hW = (const float*)d_in[4];
  const float* rna_hb = (const float*)d_in[5];
  const float* mi_W   = (const float*)d_in[6];
  const float* mi_b   = (const float*)d_in[7];
  const float* mi_hW  = (const float*)d_in[8];
  const float* mi_hb  = (const float*)d_in[9];
  const float* rlin   = (const float*)d_in[10];
  const float* mlin   = (const float*)d_in[11];
  const float* bn_g   = (const float*)d_in[12];
  const float* bn_be  = (const float*)d_in[13];
  const float* bn_m   = (const float*)d_in[14];
  const float* bn_v   = (const float*)d_in[15];
  const float* ff_W   = (const float*)d_in[16];
  const float* ff_b   = (const float*)d_in[17];
  const float* c1W    = (const float*)d_in[18];
  const float* c1b    = (const float*)d_in[19];
  const float* c2W    = (const float*)d_in[20];
  const float* c2b    = (const float*)d_in[21];
  float* out = (float*)d_out;

  char* wsp = (char*)d_ws;
  auto carve = [&](size_t bytes) -> void* {
    void* p = (void*)wsp;
    wsp += (bytes + 255) & ~(size_t)255;
    return p;
  };
  float* rna   = (float*)carve((size_t)N * 40 * 4);
  float* mi    = (float*)carve((size_t)N * 40 * 4);
  float* fused = (float*)carve((size_t)N * 60 * 4);
  float* featG = (float*)carve((size_t)N * 64 * 4);
  float* featM = (float*)carve((size_t)N * 64 * 4);
  float* featF = (float*)carve((size_t)N * 64 * 4);
  float* sqG   = (float*)carve((size_t)N * 4);
  float* sqM   = (float*)carve((size_t)N * 4);
  float* sqF   = (float*)carve((size_t)N * 4);
  int*   idxG  = (int*)carve((size_t)N * 5 * 4);
  int*   idxF  = (int*)carve((size_t)N * 3 * 4);
  int*   idxM  = (int*)carve((size_t)N * 3 * 4);
  float* wvG   = (float*)carve((size_t)N * 5 * 4);
  float* wvF   = (float*)carve((size_t)N * 3 * 4);
  float* wvM   = (float*)carve((size_t)N * 3 * 4);
  float* rsG   = (float*)carve((size_t)N * 4);
  float* rsF   = (float*)carve((size_t)N * 4);
  float* rsM   = (float*)carve((size_t)N * 4);
  float* sinv  = (float*)carve((size_t)N * 4);
  float* xb    = (float*)carve((size_t)N * 60 * 4);
  float* Xmat  = (float*)carve((size_t)N * 40 * 4);

  const dim3 tileGrid(N / 16);
  const dim3 splitBlk(32 * GNW);
  const dim3 waveBlk(32);

  // 1-2: input GEMMs with tanh (K-split across 8 waves, LDS reduce)
  gemm_bias_act<1><<<tileGrid, splitBlk, 0, stream>>>(gene, rna_W, rna_b, rna, 6000, 40);
  gemm_bias_act<1><<<tileGrid, splitBlk, 0, stream>>>(miRNA, mi_W, mi_b, mi, 600, 40);

  // 3: highway nets (in-place)
  highway_kernel<<<tileGrid, waveBlk, 0, stream>>>(rna, rna_hW, rna_hb);
  highway_kernel<<<tileGrid, waveBlk, 0, stream>>>(mi, mi_hW, mi_hb);

  // 4: fuse + L2-normalized padded features
  prep_kernel<<<(N + 127) / 128, 128, 0, stream>>>(rna, mi, rlin, mlin, fused, featG,
                                                   featM, featF, sqG, sqM, sqF, N);

  // 5: gram matrices + per-row top-k (j-split across 8 waves, LDS merge)
  gram_topk<5><<<tileGrid, splitBlk, 0, stream>>>(featG, sqG, idxG, N);
  gram_topk<3><<<tileGrid, splitBlk, 0, stream>>>(featF, sqF, idxF, N);
  gram_topk<3><<<tileGrid, splitBlk, 0, stream>>>(featM, sqM, idxM, N);

  // 6: mutual mask -> sparse weights + row/col sums
  sparse_w<5><<<(N + 127) / 128, 128, 0, stream>>>(featG, sqG, idxG, wvG, rsG, N);
  sparse_w<3><<<(N + 127) / 128, 128, 0, stream>>>(featF, sqF, idxF, wvF, rsF, N);
  sparse_w<3><<<(N + 127) / 128, 128, 0, stream>>>(featM, sqM, idxM, wvM, rsM, N);

  // 7: degree scaling
  degree_kernel<<<(N + 127) / 128, 128, 0, stream>>>(idxG, wvG, rsG, idxF, wvF, rsF,
                                                     idxM, wvM, rsM, sinv, N);

  // 8: batchnorm + feed-forward tanh GEMM -> X
  bn_kernel<<<(N * 60 + 255) / 256, 256, 0, stream>>>(fused, bn_g, bn_be, bn_m, bn_v,
                                                      xb, N * 60);
  gemm_bias_act<1><<<tileGrid, splitBlk, 0, stream>>>(xb, ff_W, ff_b, Xmat, 60, 40);

  // 9: sparse S@X + cox head
  final_kernel<<<(N + 63) / 64, 64, 0, stream>>>(Xmat, sinv, idxG, wvG, rsG, idxF, wvF,
                                                 rsF, idxM, wvM, rsM, c1W, c1b, c2W,
                                                 c2b, out, N);
}